// GatedFusion_62929860821282
// MI455X (gfx1250) — compile-verified
//
#include <hip/hip_runtime.h>
#include <math.h>

// ---------------------------------------------------------------------------
// GatedFusion, MI455X (gfx1250).
//
// Reference pruning: softmax over a singleton axis == 1.0, so the whole
// word / top-k / gather branch is dead.  Remaining math:
//   fused = sigmoid([time|text] @ gate_w^T + gate_b) * time + (1-s) * text
//   out   = fused @ out_w^T + out_b
// Two fp32 GEMMs (52.6 + 26.3 GFLOP) vs ~300 MB HBM traffic (~13 us at
// 23.3 TB/s) -> compute-bound in fp32 matrix math.  fp32 inputs => native
// V_WMMA_F32_16X16X4_F32 (no precision loss).
//
//  * 4x2 register blocking per wave (64x32): 6 fragment loads feed 8 WMMAs
//    per k-step -> lower LDS read pressure.
//  * Stage loads use GLOBAL_LOAD_ASYNC_TO_LDS_B128 (ASYNCcnt-tracked, no
//    VGPR staging).  Builtin signature (from hipcc diagnostic): args are
//    v4i addrspace(1)* global src, v4i addrspace(3)* LDS dst, imm offset,
//    imm cpol.  Manual float4 + ds_store_b128 fallback kept behind
//    __has_builtin.
// ---------------------------------------------------------------------------

typedef float v2f __attribute__((ext_vector_type(2)));
typedef float v8f __attribute__((ext_vector_type(8)));
typedef int   i4v __attribute__((ext_vector_type(4)));
typedef i4v __attribute__((address_space(1))) * g_i4p;  // global int4*
typedef i4v __attribute__((address_space(3))) * l_i4p;  // LDS int4*

#define B_SZ   16
#define T_SZ   1024
#define D_SZ   896
#define M_SZ   (B_SZ * T_SZ)   // 16384 rows
#define K1_SZ  (2 * D_SZ)      // 1792 (concat K for gate GEMM)

#define BM 128
#define BN 64
#define KB 16                   // K per LDS stage; 896 % 16 == 0 so the
                                // time|text concat boundary never splits a stage
#define LDS_LD 20               // padded row stride (floats): 20*L mod 64 all
                                // distinct for L=0..15 -> conflict-free;
                                // 80 B row pitch keeps 16B-aligned b128 stores
#define NTHREADS 128            // 4 wave32s; wave grid 2x2, each wave 64x32

#if defined(__has_builtin)
#  if __has_builtin(__builtin_amdgcn_global_load_async_to_lds_b128) && \
      __has_builtin(__builtin_amdgcn_s_wait_asynccnt)
#    define USE_ASYNC_LDS 1
#  endif
#endif
#ifndef USE_ASYNC_LDS
#  define USE_ASYNC_LDS 0
#endif

// D = A(16x4) * B(4x16) + C, all fp32.
// Per-lane fragment layout (ISA 7.12.2, 32-bit A 16x4 / B 4x16):
//   lanes 0-15 : {K=k0, K=k0+1},  lanes 16-31 : {K=k0+2, K=k0+3}
__device__ __forceinline__ v8f wmma_f32(v2f a, v2f b, v8f c) {
  return __builtin_amdgcn_wmma_f32_16x16x4_f32(
      /*neg_a=*/false, a, /*neg_b=*/false, b,
      /*c_mod=*/(short)0, c, /*reuse_a=*/false, /*reuse_b=*/false);
}

// Stage loader: ROWS x 16 floats, 128 threads, 16-byte chunks.
template <int ROWS>
__device__ __forceinline__ void stage_to_lds(float* __restrict__ dst,
                                             const float* __restrict__ src,
                                             int ld, int tid) {
  const int c4 = (tid & 3) * 4;  // column group (4 floats)
  const int r0 = tid >> 2;       // rows 0..31 per pass
#pragma unroll
  for (int rr = 0; rr < ROWS / 32; ++rr) {
    const int row = r0 + rr * 32;
#if USE_ASYNC_LDS
    __builtin_amdgcn_global_load_async_to_lds_b128(
        (g_i4p)(src + (size_t)row * ld + c4),
        (l_i4p)(dst + row * LDS_LD + c4),
        /*imm offset=*/0, /*cpol=*/0);
#else
    float4 v = *(const float4*)(src + (size_t)row * ld + c4);
    *(float4*)(dst + row * LDS_LD + c4) = v;
#endif
  }
}

// Wait for this wave's outstanding stage traffic, then block barrier.
__device__ __forceinline__ void stage_barrier() {
#if USE_ASYNC_LDS
  __builtin_amdgcn_s_wait_asynccnt(0);
#endif
  __syncthreads();
}

// 8 WMMAs on one k-step (kk = 0,4,8,12 within the stage).
__device__ __forceinline__ void mma_step(const float* __restrict__ Ab,
                                         const float* __restrict__ Bb,
                                         int mrow, int ncol, int kid, int kk,
                                         v8f acc[4][2]) {
  v2f a0 = *(const v2f*)(Ab + (mrow     ) * LDS_LD + kk + kid);
  v2f a1 = *(const v2f*)(Ab + (mrow + 16) * LDS_LD + kk + kid);
  v2f a2 = *(const v2f*)(Ab + (mrow + 32) * LDS_LD + kk + kid);
  v2f a3 = *(const v2f*)(Ab + (mrow + 48) * LDS_LD + kk + kid);
  v2f b0 = *(const v2f*)(Bb + (ncol     ) * LDS_LD + kk + kid);
  v2f b1 = *(const v2f*)(Bb + (ncol + 16) * LDS_LD + kk + kid);
  acc[0][0] = wmma_f32(a0, b0, acc[0][0]);
  acc[0][1] = wmma_f32(a0, b1, acc[0][1]);
  acc[1][0] = wmma_f32(a1, b0, acc[1][0]);
  acc[1][1] = wmma_f32(a1, b1, acc[1][1]);
  acc[2][0] = wmma_f32(a2, b0, acc[2][0]);
  acc[2][1] = wmma_f32(a2, b1, acc[2][1]);
  acc[3][0] = wmma_f32(a3, b0, acc[3][0]);
  acc[3][1] = wmma_f32(a3, b1, acc[3][1]);
}

// ---------------------------------------------------------------------------
// Kernel 1: gate logits GEMM over K=1792 (virtual concat of time|text),
// fused sigmoid + lerp epilogue -> fused[m, n] in workspace.
// ---------------------------------------------------------------------------
__global__ __launch_bounds__(NTHREADS) void gate_fuse_kernel(
    const float* __restrict__ timef, const float* __restrict__ textf,
    const float* __restrict__ gate_w,  // [D, 2D] row-major: W[n][k]
    const float* __restrict__ gate_b,  // [D]
    float* __restrict__ fused) {       // [M, D]
  __shared__ float As[2][BM * LDS_LD];
  __shared__ float Bs[2][BN * LDS_LD];

  const int tid  = threadIdx.x;
  const int lane = tid & 31;
  const int wave = tid >> 5;
  const int wr   = (wave >> 1) * 64;   // wave row offset inside 128x64 tile
  const int wc   = (wave & 1) * 32;    // wave col offset
  const int m0   = blockIdx.y * BM;
  const int n0   = blockIdx.x * BN;

  v8f acc[4][2];
  const v8f vz = {};
#pragma unroll
  for (int i = 0; i < 4; ++i) { acc[i][0] = vz; acc[i][1] = vz; }

  const int S = K1_SZ / KB;  // 112 stages

  // stage 0 (k0 = 0 -> time half)
  stage_to_lds<BM>(As[0], timef + (size_t)m0 * D_SZ, D_SZ, tid);
  stage_to_lds<BN>(Bs[0], gate_w + (size_t)n0 * K1_SZ, K1_SZ, tid);
  stage_barrier();

  const int mrow = wr + (lane & 15);
  const int ncol = wc + (lane & 15);
  const int kid  = (lane >> 4) * 2;

  for (int s = 0; s < S; ++s) {
    const int cur = s & 1;
    if (s + 1 < S) {
      const int k0 = (s + 1) * KB;
      const float* abase = (k0 < D_SZ) ? (timef + k0) : (textf + (k0 - D_SZ));
      stage_to_lds<BM>(As[cur ^ 1], abase + (size_t)m0 * D_SZ, D_SZ, tid);
      stage_to_lds<BN>(Bs[cur ^ 1], gate_w + (size_t)n0 * K1_SZ + k0, K1_SZ, tid);
    }
    const float* Ab = As[cur];
    const float* Bb = Bs[cur];
#pragma unroll
    for (int kk = 0; kk < KB; kk += 4)
      mma_step(Ab, Bb, mrow, ncol, kid, kk, acc);
    stage_barrier();
  }

  // Epilogue: C/D layout -> vgpr r, lanes<16: row r; lanes>=16: row r+8.
  const int hi = (lane >> 4) * 8;
#pragma unroll
  for (int ti = 0; ti < 4; ++ti) {
#pragma unroll
    for (int tj = 0; tj < 2; ++tj) {
      const int mb = m0 + wr + ti * 16 + hi;
      const int nc = n0 + wc + tj * 16 + (lane & 15);
      const float bias = gate_b[nc];
#pragma unroll
      for (int r = 0; r < 8; ++r) {
        const size_t idx = (size_t)(mb + r) * D_SZ + nc;
        const float logit = acc[ti][tj][r] + bias;
        const float g = 1.0f / (1.0f + __expf(-logit));
        const float tv = timef[idx];
        const float xv = textf[idx];
        fused[idx] = g * tv + (1.0f - g) * xv;
      }
    }
  }
}

// ---------------------------------------------------------------------------
// Kernel 2: out = fused @ out_w^T + out_b, K = 896.
// ---------------------------------------------------------------------------
__global__ __launch_bounds__(NTHREADS) void out_gemm_kernel(
    const float* __restrict__ fused,   // [M, D]
    const float* __restrict__ out_w,   // [D, D] row-major: W[n][k]
    const float* __restrict__ out_b,   // [D]
    float* __restrict__ out) {         // [M, D]
  __shared__ float As[2][BM * LDS_LD];
  __shared__ float Bs[2][BN * LDS_LD];

  const int tid  = threadIdx.x;
  const int lane = tid & 31;
  const int wave = tid >> 5;
  const int wr   = (wave >> 1) * 64;
  const int wc   = (wave & 1) * 32;
  const int m0   = blockIdx.y * BM;
  const int n0   = blockIdx.x * BN;

  v8f acc[4][2];
  const v8f vz = {};
#pragma unroll
  for (int i = 0; i < 4; ++i) { acc[i][0] = vz; acc[i][1] = vz; }

  const int S = D_SZ / KB;  // 56 stages

  stage_to_lds<BM>(As[0], fused + (size_t)m0 * D_SZ, D_SZ, tid);
  stage_to_lds<BN>(Bs[0], out_w + (size_t)n0 * D_SZ, D_SZ, tid);
  stage_barrier();

  const int mrow = wr + (lane & 15);
  const int ncol = wc + (lane & 15);
  const int kid  = (lane >> 4) * 2;

  for (int s = 0; s < S; ++s) {
    const int cur = s & 1;
    if (s + 1 < S) {
      const int k0 = (s + 1) * KB;
      stage_to_lds<BM>(As[cur ^ 1], fused + (size_t)m0 * D_SZ + k0, D_SZ, tid);
      stage_to_lds<BN>(Bs[cur ^ 1], out_w + (size_t)n0 * D_SZ + k0, D_SZ, tid);
    }
    const float* Ab = As[cur];
    const float* Bb = Bs[cur];
#pragma unroll
    for (int kk = 0; kk < KB; kk += 4)
      mma_step(Ab, Bb, mrow, ncol, kid, kk, acc);
    stage_barrier();
  }

  const int hi = (lane >> 4) * 8;
#pragma unroll
  for (int ti = 0; ti < 4; ++ti) {
#pragma unroll
    for (int tj = 0; tj < 2; ++tj) {
      const int mb = m0 + wr + ti * 16 + hi;
      const int nc = n0 + wc + tj * 16 + (lane & 15);
      const float bias = out_b[nc];
#pragma unroll
      for (int r = 0; r < 8; ++r) {
        out[(size_t)(mb + r) * D_SZ + nc] = acc[ti][tj][r] + bias;
      }
    }
  }
}

// ---------------------------------------------------------------------------
extern "C" void kernel_launch(void* const* d_in, const int* in_sizes, int n_in,
                              void* d_out, int out_size, void* d_ws, size_t ws_size,
                              hipStream_t stream) {
  const float* timef  = (const float*)d_in[0];   // [B,T,D]
  const float* textf  = (const float*)d_in[1];   // [B,T,D]
  // d_in[2] = word: DEAD (softmax over singleton axis == 1.0)
  const float* gate_w = (const float*)d_in[3];   // [D, 2D]
  const float* gate_b = (const float*)d_in[4];   // [D]
  const float* out_w  = (const float*)d_in[5];   // [D, D]
  const float* out_b  = (const float*)d_in[6];   // [D]
  // d_in[7] = s: unused

  float* fusedp = (float*)d_ws;   // needs M_SZ * D_SZ * 4 = 58.7 MB scratch
  float* outp   = (float*)d_out;  // [B,T,D] fp32

  dim3 grid(D_SZ / BN, M_SZ / BM);   // (14, 128)
  dim3 block(NTHREADS);

  gate_fuse_kernel<<<grid, block, 0, stream>>>(timef, textf, gate_w, gate_b, fusedp);
  out_gemm_kernel<<<grid, block, 0, stream>>>(fusedp, out_w, out_b, outp);
}